// GATLayer_20770461843679
// MI455X (gfx1250) — compile-verified
//
#include <hip/hip_runtime.h>
#include <math.h>

// Problem constants from the reference (ND, DIN, FATTN, SLOPE are compile-time
// constants of the model; N and E are taken from in_sizes).
#define DIN   512
#define FATTN 256
#define NDIS  4000
#define SLOPE 0.2f

typedef float v2f __attribute__((ext_vector_type(2)));
typedef float v8f __attribute__((ext_vector_type(8)));

// Order-preserving monotone encoding of fp32 into uint for atomicMax.
__device__ __forceinline__ unsigned enc_f32(float x) {
  unsigned u = __float_as_uint(x);
  return (u & 0x80000000u) ? ~u : (u | 0x80000000u);
}
__device__ __forceinline__ float dec_f32(unsigned k) {
  return (k & 0x80000000u) ? __uint_as_float(k & 0x7FFFFFFFu)
                           : __uint_as_float(~k);
}

// ---------------------------------------------------------------------------
// Init: zero h (= d_out) and denom, set segment-max keys to enc(-inf).
// ---------------------------------------------------------------------------
__global__ void init_kernel(float* __restrict__ h, unsigned* __restrict__ keys,
                            float* __restrict__ denom, int n_h, int n_nodes) {
  int i = blockIdx.x * blockDim.x + threadIdx.x;
  if (i < n_h) h[i] = 0.0f;
  if (i < n_nodes) {
    keys[i]  = 0x007FFFFFu;  // enc(-inf)
    denom[i] = 0.0f;
  }
}

// ---------------------------------------------------------------------------
// Projection GEMM with fp32 WMMA (V_WMMA_F32_16X16X4_F32).
// z[n,f] = sum_k X[n,k] * W[f,k], where (X,W) = (d_sim,Wd) for n<ND else
// (me_sim,Wm). ND is 16-aligned, so each 16-row M-tile is homogeneous.
// One wave computes a 16x64 output strip: 4 accumulators share one A tile.
//
// ISA layouts (wave32):
//   A 16x4  f32: M = lane%16, k = 2*(lane/16) + vgpr   -> b64 load per step
//   B 4x16  f32: N = lane%16, k = 2*(lane/16) + vgpr   -> b64 load per step
//   C 16x16 f32: M = 8*(lane/16) + vgpr, N = lane%16
// ---------------------------------------------------------------------------
__global__ void __launch_bounds__(128)
proj_wmma_kernel(const float* __restrict__ d_sim, const float* __restrict__ me_sim,
                 const float* __restrict__ Wd,    const float* __restrict__ Wm,
                 float* __restrict__ z) {
  const int lane  = threadIdx.x & 31;
  const int wave  = threadIdx.x >> 5;   // 0..3 -> F-group of 64
  const int mtile = blockIdx.x;         // 16 nodes per tile
  const int fbase = wave * 64;
  const int l15   = lane & 15;
  const int hk2   = (lane >> 4) * 2;    // k sub-offset: 0 or 2

  const bool dis = (mtile * 16) < NDIS;
  const float* __restrict__ X = dis ? d_sim : me_sim;
  const float* __restrict__ W = dis ? Wd : Wm;

  const float* aptr = X + (size_t)(mtile * 16 + l15) * DIN + hk2;
  const float* bptr = W + (size_t)(fbase + l15) * DIN + hk2;

  v8f acc0 = {}, acc1 = {}, acc2 = {}, acc3 = {};

  for (int k = 0; k < DIN; k += 4) {
    v2f a  = *(const v2f*)(aptr + k);
    v2f b0 = *(const v2f*)(bptr + k);
    v2f b1 = *(const v2f*)(bptr + 16 * DIN + k);
    v2f b2 = *(const v2f*)(bptr + 32 * DIN + k);
    v2f b3 = *(const v2f*)(bptr + 48 * DIN + k);
    acc0 = __builtin_amdgcn_wmma_f32_16x16x4_f32(false, a, false, b0, (short)0, acc0, false, false);
    acc1 = __builtin_amdgcn_wmma_f32_16x16x4_f32(false, a, false, b1, (short)0, acc1, false, false);
    acc2 = __builtin_amdgcn_wmma_f32_16x16x4_f32(false, a, false, b2, (short)0, acc2, false, false);
    acc3 = __builtin_amdgcn_wmma_f32_16x16x4_f32(false, a, false, b3, (short)0, acc3, false, false);
  }

  const int row0 = mtile * 16 + (lane >> 4) * 8;
#pragma unroll
  for (int v = 0; v < 8; ++v) {
    float* zr = z + (size_t)(row0 + v) * FATTN + fbase + l15;
    zr[0]  = acc0[v];
    zr[16] = acc1[v];
    zr[32] = acc2[v];
    zr[48] = acc3[v];
  }
}

// ---------------------------------------------------------------------------
// Per-edge attention logit: e = leaky_relu(dot(z[src], z[dst])), and
// segment-max over dst via monotone-encoded atomicMax. One wave per edge,
// each lane reduces 8 of the 256 features (two aligned float4 loads).
// ---------------------------------------------------------------------------
__global__ void __launch_bounds__(256)
edge_logit_kernel(const float* __restrict__ z, const int* __restrict__ src,
                  const int* __restrict__ dst, float* __restrict__ e,
                  unsigned* __restrict__ keys, int nE) {
  const int lane = threadIdx.x & 31;
  const int edge = blockIdx.x * 8 + (threadIdx.x >> 5);
  if (edge >= nE) return;
  const int s = src[edge], d = dst[edge];
  const float4* zs = (const float4*)(z + (size_t)s * FATTN) + lane * 2;
  const float4* zd = (const float4*)(z + (size_t)d * FATTN) + lane * 2;
  float4 a0 = zs[0], a1 = zs[1];
  float4 b0 = zd[0], b1 = zd[1];
  float sum = a0.x * b0.x + a0.y * b0.y + a0.z * b0.z + a0.w * b0.w
            + a1.x * b1.x + a1.y * b1.y + a1.z * b1.z + a1.w * b1.w;
#pragma unroll
  for (int off = 16; off > 0; off >>= 1) sum += __shfl_xor(sum, off, 32);
  if (lane == 0) {
    float lr = sum > 0.0f ? sum : SLOPE * sum;
    e[edge] = lr;
    atomicMax(keys + d, enc_f32(lr));
  }
}

// ---------------------------------------------------------------------------
// ee = exp(e - emax[dst]); denom[dst] += ee.  One thread per edge.
// ---------------------------------------------------------------------------
__global__ void edge_exp_kernel(const float* __restrict__ e, const int* __restrict__ dst,
                                const unsigned* __restrict__ keys,
                                float* __restrict__ ee, float* __restrict__ denom, int nE) {
  int i = blockIdx.x * blockDim.x + threadIdx.x;
  if (i >= nE) return;
  const int d = dst[i];
  float m = dec_f32(keys[d]);
  if (!isfinite(m)) m = 0.0f;
  float x = expf(e[i] - m);
  ee[i] = x;
  atomicAdd(denom + d, x);
}

// ---------------------------------------------------------------------------
// h[dst] += (ee/denom) * z[src].  One wave per edge, 8 features per lane,
// float atomic adds resolved in L2 (z and h are L2-resident: ~20 MB total).
// ---------------------------------------------------------------------------
__global__ void __launch_bounds__(256)
edge_agg_kernel(const float* __restrict__ z, const int* __restrict__ src,
                const int* __restrict__ dst, const float* __restrict__ ee,
                const float* __restrict__ denom, float* __restrict__ h, int nE) {
  const int lane = threadIdx.x & 31;
  const int edge = blockIdx.x * 8 + (threadIdx.x >> 5);
  if (edge >= nE) return;
  const int s = src[edge], d = dst[edge];
  const float alpha = ee[edge] / fmaxf(denom[d], 1e-9f);
  const float* zs = z + (size_t)s * FATTN + lane * 8;
  float*       hd = h + (size_t)d * FATTN + lane * 8;
#pragma unroll
  for (int j = 0; j < 8; ++j) atomicAdd(hd + j, alpha * zs[j]);
}

// ---------------------------------------------------------------------------
// In-place ELU on h (= d_out).
// ---------------------------------------------------------------------------
__global__ void elu_kernel(float* __restrict__ h, int n) {
  int i = blockIdx.x * blockDim.x + threadIdx.x;
  if (i < n) {
    float x = h[i];
    h[i] = x > 0.0f ? x : (expf(x) - 1.0f);
  }
}

// ---------------------------------------------------------------------------
extern "C" void kernel_launch(void* const* d_in, const int* in_sizes, int n_in,
                              void* d_out, int out_size, void* d_ws, size_t ws_size,
                              hipStream_t stream) {
  const float* d_sim  = (const float*)d_in[0];
  const float* me_sim = (const float*)d_in[1];
  const float* Wd     = (const float*)d_in[2];
  const float* Wm     = (const float*)d_in[3];
  const int*   src    = (const int*)d_in[4];
  const int*   dst    = (const int*)d_in[5];

  const int N  = in_sizes[0] / DIN;   // 10000
  const int nE = in_sizes[4];         // 320000
  float* out = (float*)d_out;         // doubles as h accumulator

  // Workspace layout (~13 MB): z | e | ee | keys | denom
  char* ws = (char*)d_ws;
  float*    z     = (float*)ws;              size_t off = (size_t)N * FATTN * sizeof(float);
  float*    e     = (float*)(ws + off);      off += (size_t)nE * sizeof(float);
  float*    ee    = (float*)(ws + off);      off += (size_t)nE * sizeof(float);
  unsigned* keys  = (unsigned*)(ws + off);   off += (size_t)N * sizeof(unsigned);
  float*    denom = (float*)(ws + off);

  const int nh = N * FATTN;

  init_kernel<<<(nh + 255) / 256, 256, 0, stream>>>(out, keys, denom, nh, N);
  proj_wmma_kernel<<<N / 16, 128, 0, stream>>>(d_sim, me_sim, Wd, Wm, z);
  edge_logit_kernel<<<(nE + 7) / 8, 256, 0, stream>>>(z, src, dst, e, keys, nE);
  edge_exp_kernel<<<(nE + 255) / 256, 256, 0, stream>>>(e, dst, keys, ee, denom, nE);
  edge_agg_kernel<<<(nE + 7) / 8, 256, 0, stream>>>(z, src, dst, ee, denom, out, nE);
  elu_kernel<<<(nh + 255) / 256, 256, 0, stream>>>(out, nh);
}